// CRsAERandProj_47029891891290
// MI455X (gfx1250) — compile-verified
//
#include <hip/hip_runtime.h>
#include <math.h>

typedef __attribute__((ext_vector_type(2))) float v2f;
typedef __attribute__((ext_vector_type(8))) float v8f;

namespace {
constexpr int kDIn   = 64;
constexpr int kDOrg  = 512;
constexpr int kDEnc  = 1024;
constexpr int kRows  = 16;                 // batch rows per workgroup (one WMMA M-tile)
constexpr int kWaves = 8;                  // waves per workgroup (256 threads)
constexpr int kThreads = kWaves * 32;
constexpr int kKSpan = kDEnc / kWaves;     // enc columns owned per wave = 128
constexpr int kChunks = kKSpan / 16;       // 16-wide N chunks per wave = 8
constexpr float kInvL = 1.0f / 50.0f;
constexpr float kThr  = 0.1f / 50.0f;
} // namespace

__device__ __forceinline__ v8f wmma_f32(v2f a, v2f b, v8f c) {
  // V_WMMA_F32_16X16X4_F32 : full-precision fp32 matrix core op (wave32)
  return __builtin_amdgcn_wmma_f32_16x16x4_f32(false, a, false, b, (short)0, c,
                                               false, false);
}

__device__ __forceinline__ v8f zero8() {
  v8f z;
#pragma unroll
  for (int i = 0; i < 8; ++i) z[i] = 0.0f;
  return z;
}

// proj[16][64] += src[16][1024](LDS) x phiH^T   (this wave's 128-wide K slice)
// B1[k][n] = phiH[n][k] -> consecutive k are contiguous in phiH row-major.
__device__ __forceinline__ void proj_accum(const float* src, const float* phiH,
                                           float* proj, int wave, int mrow,
                                           int half) {
  const int kbase = wave * kKSpan;
  const float* arow = src + mrow * kDEnc + kbase + 2 * half;   // A: row = lane%16
  const float* brow = phiH + mrow * kDEnc + kbase + 2 * half;  // B: col n = lane%16 (+16/chunk)
  v8f acc[4] = {zero8(), zero8(), zero8(), zero8()};
  for (int ks = 0; ks < kKSpan / 4; ++ks) {
    v2f a = *(const v2f*)(arow + 4 * ks);
#pragma unroll
    for (int c = 0; c < 4; ++c) {
      v2f b = *(const v2f*)(brow + c * 16 * kDEnc + 4 * ks);
      acc[c] = wmma_f32(a, b, acc[c]);
    }
  }
#pragma unroll
  for (int c = 0; c < 4; ++c)
#pragma unroll
    for (int v = 0; v < 8; ++v)
      atomicAdd(&proj[(v + 8 * half) * kDIn + c * 16 + mrow], acc[c][v]);
}

// phiH = (phi @ H)  [64,1024]  (also store transpose for step-2 B fragments)
__global__ void prep_phiH(const float* __restrict__ H,
                          const float* __restrict__ phi,
                          float* __restrict__ phiH, float* __restrict__ phiHT) {
  const int idx = blockIdx.x * blockDim.x + threadIdx.x;
  const int i = idx >> 10;          // 0..63   (D_in)
  const int e = idx & (kDEnc - 1);  // 0..1023 (D_enc)
  const float* ph = phi + i * kDOrg;
  float s = 0.0f;
#pragma unroll 8
  for (int o = 0; o < kDOrg; ++o) s = fmaf(ph[o], H[o * kDEnc + e], s);
  phiH[i * kDEnc + e] = s;
  phiHT[e * kDIn + i] = s;
}

__global__ void __launch_bounds__(kThreads)
fista_kernel(const float* __restrict__ x, const float* __restrict__ phiH,
             const float* __restrict__ phiHT, const int* __restrict__ Tptr,
             float* __restrict__ out, int B) {
  extern __shared__ float smem[];
  float* yk   = smem;                   // [16][1024]  FISTA y state (resident)
  float* proj = smem + kRows * kDEnc;   // [16][64]    yk @ phiH^T partial sums
  float* x2s  = proj + kRows * kDIn;    // [16][64]    input tile

  const int T    = *Tptr;
  const int row0 = blockIdx.x * kRows;
  const int tid  = threadIdx.x;
  const int lane = tid & 31;
  const int wave = tid >> 5;
  const int mrow = lane & 15;
  const int half = lane >> 4;

  for (int i = tid; i < kRows * kDEnc; i += kThreads) yk[i] = 0.0f;
  for (int i = tid; i < kRows * kDIn; i += kThreads) {
    proj[i] = 0.0f;
    x2s[i]  = x[(row0 + (i >> 6)) * kDIn + (i & 63)];
  }
  __syncthreads();

  // x_old kept in registers in exact C-fragment layout (8 chunks x 8 VGPRs)
  float xold[kChunks][8];
#pragma unroll
  for (int c = 0; c < kChunks; ++c)
#pragma unroll
    for (int v = 0; v < 8; ++v) xold[c][v] = 0.0f;

  float t = 1.0f;

  for (int it = 0; it < T; ++it) {
    // ---- step 1: proj = yk @ phiH^T (cross-wave K split, LDS float atomics)
    proj_accum(yk, phiH, proj, wave, mrow, half);
    __syncthreads();

    // ---- packed resid A-fragments for all 16 k-steps of step 2 (K = 64)
    v2f rfrag[16];
#pragma unroll
    for (int j = 0; j < 16; ++j) {
      const int k = 4 * j + 2 * half;
      v2f xv = *(const v2f*)(x2s + mrow * kDIn + k);
      v2f pv = *(const v2f*)(proj + mrow * kDIn + k);
      rfrag[j] = xv - pv;
    }
    __syncthreads();
    for (int i = tid; i < kRows * kDIn; i += kThreads) proj[i] = 0.0f;

    const float tn  = 0.5f * (1.0f + sqrtf(1.0f + 4.0f * t * t));
    const float mom = (t - 1.0f) / tn;
    t = tn;

    // ---- step 2: x_new = soft(yk + resid @ phiH / L); yk = x_new + mom*dx
    // Each wave owns enc columns [wave*128, wave*128+128): no cross-wave races.
#pragma unroll
    for (int c = 0; c < kChunks; ++c) {
      const int ng = wave * kKSpan + c * 16 + mrow;   // global enc column
      const float* bt = phiHT + ng * kDIn + 2 * half; // B2[k][n]=phiH[k][n]
      v8f acc = zero8();
#pragma unroll
      for (int j = 0; j < 16; ++j) {
        v2f b = *(const v2f*)(bt + 4 * j);
        acc = wmma_f32(rfrag[j], b, acc);
      }
#pragma unroll
      for (int v = 0; v < 8; ++v) {
        const int m = v + 8 * half;
        const float ykv = yk[m * kDEnc + ng];
        float xn = fmaf(acc[v], kInvL, ykv);
        const float s = fabsf(xn) - kThr;
        xn = (s > 0.0f) ? copysignf(s, xn) : 0.0f;       // soft threshold
        const float yn = fmaf(mom, xn - xold[c][v], xn); // momentum
        xold[c][v] = xn;
        yk[m * kDEnc + ng] = yn;
      }
    }
    __syncthreads();
  }

  // ---- stash x_new into the yk LDS buffer for the final projection + output
#pragma unroll
  for (int c = 0; c < kChunks; ++c) {
    const int ng = wave * kKSpan + c * 16 + mrow;
#pragma unroll
    for (int v = 0; v < 8; ++v) yk[(v + 8 * half) * kDEnc + ng] = xold[c][v];
  }
  __syncthreads();

  // z = x_new @ phiH^T  (proj was zeroed inside the last iteration)
  proj_accum(yk, phiH, proj, wave, mrow, half);
  __syncthreads();

  // outputs: z [B,64,1] flat first, then x_new [B,1024,1]
  for (int i = tid; i < kRows * kDIn; i += kThreads)
    out[(row0 + (i >> 6)) * kDIn + (i & 63)] = proj[i];

  float* xout = out + (size_t)B * kDIn;
  for (int i = tid; i < kRows * kDEnc / 4; i += kThreads) {
    const int r  = i >> 8;
    const int c4 = i & 255;
    ((float4*)(xout + (size_t)(row0 + r) * kDEnc))[c4] =
        ((const float4*)(yk + r * kDEnc))[c4];
  }
}

extern "C" void kernel_launch(void* const* d_in, const int* in_sizes, int n_in,
                              void* d_out, int out_size, void* d_ws,
                              size_t ws_size, hipStream_t stream) {
  const float* x   = (const float*)d_in[0];
  const float* H   = (const float*)d_in[1];
  const float* phi = (const float*)d_in[2];
  const int*   T   = (const int*)d_in[3];
  float* out = (float*)d_out;
  const int B = in_sizes[0] / kDIn;

  float* phiH  = (float*)d_ws;            // [64][1024]
  float* phiHT = phiH + kDIn * kDEnc;     // [1024][64]

  prep_phiH<<<(kDIn * kDEnc) / 256, 256, 0, stream>>>(H, phi, phiH, phiHT);

  const size_t shmem =
      (size_t)(kRows * kDEnc + 2 * kRows * kDIn) * sizeof(float); // 73728 B
  (void)hipFuncSetAttribute((const void*)fista_kernel,
                            hipFuncAttributeMaxDynamicSharedMemorySize,
                            (int)shmem);
  fista_kernel<<<B / kRows, kThreads, shmem, stream>>>(x, phiH, phiHT, T, out,
                                                       B);
}